// GNNEncoder_70497593197361
// MI455X (gfx1250) — compile-verified
//
#include <hip/hip_runtime.h>
#include <hip/hip_bf16.h>

typedef float v2f __attribute__((ext_vector_type(2)));
typedef float v8f __attribute__((ext_vector_type(8)));

#define K_DIM 256
#define C_DIM 128

// ---------------- degree init: 2 self-loop copies per node ----------------
__global__ void k_init_deg(float* __restrict__ deg, int N) {
    int i = blockIdx.x * blockDim.x + threadIdx.x;
    if (i < N) deg[i] = 2.0f;
}

// ---------------- degree accumulate over edges (col) ----------------------
__global__ void k_degree(const int* __restrict__ col, float* __restrict__ deg, int E) {
    int e = blockIdx.x * blockDim.x + threadIdx.x;
    if (e < E) atomicAdd(&deg[col[e]], 1.0f);
}

// ---------------- deg -> deg^{-1/2} in place ------------------------------
__global__ void k_rsqrt(float* __restrict__ deg, int N) {
    int i = blockIdx.x * blockDim.x + threadIdx.x;
    if (i < N) {
        float d = deg[i];
        deg[i] = (d > 0.0f) ? rsqrtf(d) : 0.0f;
    }
}

// ---------------- h = x @ W via V_WMMA_F32_16X16X4_F32 --------------------
// One wave computes one 16x16 tile of h. Block = 256 threads = 8 waves
// covering a 16-row x 128-col stripe. K loop: 64 WMMAs of K=4 each.
__global__ __launch_bounds__(256) void k_gemm(const float* __restrict__ x,
                                              const float* __restrict__ W,
                                              float* __restrict__ h, int N) {
    const int wave = threadIdx.x >> 5;   // 0..7 -> N-tile
    const int lane = threadIdx.x & 31;
    const int lm   = lane & 15;          // M (for A) / N (for B,C) within tile
    const int hi   = lane >> 4;          // lane-half selector
    const int mbase = blockIdx.x << 4;
    const int nbase = wave << 4;

    int mrow = mbase + lm;
    if (mrow >= N) mrow = N - 1;         // branchless clamp (keeps EXEC full)
    const float* xrow = x + (size_t)mrow * K_DIM;

    v8f c = {};
    #pragma unroll 4
    for (int k0 = 0; k0 < K_DIM; k0 += 4) {
        // A 16x4 f32: lanes 0-15 hold K=k0,k0+1 ; lanes 16-31 hold K=k0+2,k0+3
        const int ka = k0 + 2 * hi;
        v2f a;
        a.x = xrow[ka];
        a.y = xrow[ka + 1];
        // B 4x16 f32: VGPR0 = row K=ka, VGPR1 = row K=ka+1, col = nbase+lm
        v2f bmat;
        bmat.x = W[(size_t)ka * C_DIM + nbase + lm];
        bmat.y = W[(size_t)(ka + 1) * C_DIM + nbase + lm];
        c = __builtin_amdgcn_wmma_f32_16x16x4_f32(
                /*neg_a=*/false, a, /*neg_b=*/false, bmat,
                /*c_mod=*/(short)0, c, /*reuse_a=*/false, /*reuse_b=*/false);
    }

    // C/D layout: VGPR r -> M=r (lanes 0-15), M=r+8 (lanes 16-31); N = lm
    float* hbase = h + (size_t)(mbase + (hi << 3)) * C_DIM + nbase + lm;
    if (mbase + 16 <= N) {
        // fast path: whole tile in range, no per-row predication
        #pragma unroll
        for (int r = 0; r < 8; ++r) {
            hbase[(size_t)r * C_DIM] = c[r];
        }
    } else {
        #pragma unroll
        for (int r = 0; r < 8; ++r) {
            int m = mbase + r + (hi << 3);
            if (m < N) h[(size_t)m * C_DIM + nbase + lm] = c[r];
        }
    }
}

// ------- seed accumulator with self-loop messages: out = 2*dis^2*h --------
__global__ void k_selfmsg(const float* __restrict__ h, const float* __restrict__ dis,
                          float* __restrict__ out, long long total) {
    long long i = blockIdx.x * (long long)blockDim.x + threadIdx.x;
    if (i >= total) return;
    int node = (int)(i >> 7);
    float d = dis[node];
    out[i] = 2.0f * d * d * h[i];
}

// ------- edge scatter: one wave per edge, float4 gather + 4 f32 atomics ---
// Edge index forced uniform (SGPR) so row/col/dis loads scalarize and the
// gather/atomics use SGPR-base + VGPR-offset addressing.
__global__ __launch_bounds__(256) void k_scatter(const int* __restrict__ row,
                                                 const int* __restrict__ col,
                                                 const float* __restrict__ dis,
                                                 const float* __restrict__ h,
                                                 float* __restrict__ out, int E) {
    int e = __builtin_amdgcn_readfirstlane((int)(blockIdx.x * 8u) + (int)(threadIdx.x >> 5));
    int lane = threadIdx.x & 31;
    if (e >= E) return;
    int r = row[e];
    int c = col[e];
    float nrm = dis[r] * dis[c];
    float4 v = ((const float4*)(h + (size_t)r * C_DIM))[lane];
    float* o = out + (size_t)c * C_DIM + lane * 4;
    atomicAdd(o + 0, v.x * nrm);
    atomicAdd(o + 1, v.y * nrm);
    atomicAdd(o + 2, v.z * nrm);
    atomicAdd(o + 3, v.w * nrm);
}

// ------- bias + leakyReLU(0.2) + LayerNorm, one wave per node, in place ---
__global__ __launch_bounds__(256) void k_finalize(float* __restrict__ out,
                                                  const float* __restrict__ bias,
                                                  const float* __restrict__ gamma,
                                                  const float* __restrict__ beta, int N) {
    int node = __builtin_amdgcn_readfirstlane((int)(blockIdx.x * 8u) + (int)(threadIdx.x >> 5));
    int lane = threadIdx.x & 31;
    if (node >= N) return;

    float* rowp = out + (size_t)node * C_DIM;
    float4 v  = ((const float4*)rowp)[lane];
    float4 bb = ((const float4*)bias)[lane];
    v.x += bb.x; v.y += bb.y; v.z += bb.z; v.w += bb.w;

    v.x = (v.x > 0.0f) ? v.x : 0.2f * v.x;
    v.y = (v.y > 0.0f) ? v.y : 0.2f * v.y;
    v.z = (v.z > 0.0f) ? v.z : 0.2f * v.z;
    v.w = (v.w > 0.0f) ? v.w : 0.2f * v.w;

    float s  = v.x + v.y + v.z + v.w;
    float s2 = v.x * v.x + v.y * v.y + v.z * v.z + v.w * v.w;
    #pragma unroll
    for (int m = 16; m >= 1; m >>= 1) {
        s  += __shfl_xor(s,  m, 32);
        s2 += __shfl_xor(s2, m, 32);
    }
    float mu   = s * (1.0f / 128.0f);
    float var  = s2 * (1.0f / 128.0f) - mu * mu;
    float rstd = rsqrtf(var + 1e-5f);

    float4 g  = ((const float4*)gamma)[lane];
    float4 be = ((const float4*)beta)[lane];
    v.x = (v.x - mu) * rstd * g.x + be.x;
    v.y = (v.y - mu) * rstd * g.y + be.y;
    v.z = (v.z - mu) * rstd * g.z + be.z;
    v.w = (v.w - mu) * rstd * g.w + be.w;

    ((float4*)rowp)[lane] = v;
}

extern "C" void kernel_launch(void* const* d_in, const int* in_sizes, int n_in,
                              void* d_out, int out_size, void* d_ws, size_t ws_size,
                              hipStream_t stream) {
    const float* x     = (const float*)d_in[0];
    const int*   ei    = (const int*)d_in[1];   // int32 (JAX default x64-disabled)
    const float* W     = (const float*)d_in[2];
    const float* bias  = (const float*)d_in[3];
    const float* gamma = (const float*)d_in[4];
    const float* beta  = (const float*)d_in[5];

    const int N = in_sizes[0] / K_DIM;   // 100000
    const int E = in_sizes[1] / 2;       // 1600000
    const int* row = ei;
    const int* col = ei + E;

    float* h   = (float*)d_ws;                       // N*128 floats (51.2 MB)
    float* deg = h + (size_t)N * C_DIM;              // N floats -> becomes deg^{-1/2}
    float* out = (float*)d_out;                      // accumulator, then final

    k_init_deg<<<(N + 255) / 256, 256, 0, stream>>>(deg, N);
    k_degree  <<<(E + 255) / 256, 256, 0, stream>>>(col, deg, E);
    k_rsqrt   <<<(N + 255) / 256, 256, 0, stream>>>(deg, N);

    k_gemm    <<<(N + 15) / 16, 256, 0, stream>>>(x, W, h, N);

    long long tot = (long long)N * C_DIM;
    k_selfmsg <<<(int)((tot + 255) / 256), 256, 0, stream>>>(h, deg, out, tot);

    k_scatter <<<(E + 7) / 8, 256, 0, stream>>>(row, col, deg, h, out, E);

    k_finalize<<<(N + 7) / 8, 256, 0, stream>>>(out, bias, gamma, beta, N);
}